// GaussianBlur2D_79216376808098
// MI455X (gfx1250) — compile-verified
//
#include <hip/hip_runtime.h>

// Gaussian blur via separable banded circular convolution, cast as
// banded-Toeplitz matmuls on V_WMMA_F32_16X16X4_F32 (full fp32 precision).
//
// Roofline (MI455X): 2 passes x (read+write 128 MiB) = 512 MiB -> ~22 us at
// 23.3 TB/s (the 128 MiB intermediate even fits the 192 MB L2). Direct VALU
// fp32 would be ~200 us compute-bound; routing the 17.3 GFLOP through the
// f32 WMMA pipe (9.4M wmma_16x16x4 ops) keeps the kernel memory-bound.
//
// v4: LDS staging now uses the CDNA5 direct memory->LDS async path
// (GLOBAL_LOAD_ASYNC_TO_LDS_B128, ASYNCcnt) instead of the
// global_load->VGPR->ds_store round trip; completion via s_wait_asynccnt 0
// before the workgroup barrier.

#define Wd    2048
#define Hd    2048
#define NB    8
#define MASKW 2047
#define RAD   64                 // truncation radius; exact in fp32 for sigma <= ~10
#define TAPS  (2*RAD + 1)        // 129
#define TILE_K (2*RAD + 16)      // 144 = K extent per 16-wide output tile
#define WTSZ  160                // reversed table: band at [15,143], zeros elsewhere
#define WCTR  79                 // rwt[j] = G(j - WCTR)
#define ROWP  260                // row-pass LDS pitch (dwords): bank = (4m + 2hi) -> conflict-free b64
#define COLP  136                // col-pass LDS pitch (dwords): lo/hi lane halves -> disjoint banks

typedef __attribute__((ext_vector_type(2))) float v2f;
typedef __attribute__((ext_vector_type(8))) float v8f;

__device__ __forceinline__ float sigma_from(const float* sigp) {
    float s = sigp[0];
    return fmaxf(s, 0.0f) + 1e-6f;   // relu(sigma_) + 1e-6
}

// Async direct copy of 16 bytes: global (SADDR base + 32-bit voff) -> LDS at
// per-lane byte address.  Tracked by ASYNCcnt.
__device__ __forceinline__ void async_g2l_b128(void* lds_dst, const void* sbase,
                                               unsigned goff_bytes) {
    unsigned ldsaddr = (unsigned)(uintptr_t)lds_dst;
    asm volatile("global_load_async_to_lds_b128 %0, %1, %2"
                 :: "v"(ldsaddr), "v"(goff_bytes), "s"(sbase)
                 : "memory");
}

__device__ __forceinline__ void wait_async_then_barrier() {
    asm volatile("s_wait_asynccnt 0x0" ::: "memory");
    __syncthreads();
}

// Reversed zero-padded tap table: rwt[j] = exp(-(j-79)^2/(2 sigma^2)) for
// |j-79| <= RAD (j in [15,143]), else 0.  Gaussian symmetry makes the
// reversal value-preserving; it flips index direction so weight pairs are
// fetched at ascending addresses.
__device__ __forceinline__ void fill_taps(float* rwt, int tid, float inv2s2) {
    if (tid < WTSZ) {
        float v = 0.0f;
        int d = tid - WCTR;
        if (d >= -RAD && d <= RAD) {
            float fd = (float)d;
            v = __expf(-fd * fd * inv2s2);
        }
        rwt[tid] = v;
    }
}

// Wave-level sum of the padded table (padding zeros are harmless).
__device__ __forceinline__ float tap_sum(const float* rwt, int lane) {
    float ps = 0.0f;
    #pragma unroll
    for (int i = 0; i < WTSZ / 32; ++i) ps += rwt[lane + 32 * i];
    #pragma unroll
    for (int off = 16; off > 0; off >>= 1) ps += __shfl_xor(ps, off, 32);
    return ps;
}

// ---------------- Pass 1: horizontal circular convolution ----------------
// Block: 256 threads (8 waves), output tile 16 rows x 128 cols.
// Per wave: one 16x16 tile.  O = A(16xK data) * B(Kx16 Toeplitz weights).
__global__ __launch_bounds__(256) void blur_row_pass(const float* __restrict__ x,
                                                     const float* __restrict__ sigp,
                                                     float* __restrict__ t) {
    __shared__ __align__(16) float rwt[WTSZ];
    __shared__ __align__(16) float lx[16 * ROWP];   // 16 rows x 256 cols (+pad)

    const int tid = threadIdx.x;
    const int b0  = blockIdx.x * 128;   // first output column
    const int a0  = blockIdx.y * 16;    // first output row
    const int bz  = blockIdx.z;

    // stage 16 x 256 input strip (column-wrapped), async global->LDS b128
    const float* xb = x + (size_t)(bz * Hd + a0) * Wd;   // uniform block base
    #pragma unroll
    for (int it = 0; it < 4; ++it) {
        int idx4 = tid + it * 256;          // 16*64 float4 total
        int row  = idx4 >> 6;
        int c4   = (idx4 & 63) << 2;
        int gc   = (b0 - RAD + c4) & MASKW; // b0-RAD is 4-aligned; no straddle at wrap
        async_g2l_b128(&lx[row * ROWP + c4], xb,
                       (unsigned)((row * Wd + gc) * 4));
    }

    const float sig = sigma_from(sigp);
    const float inv2s2 = 1.0f / (2.0f * sig * sig);
    fill_taps(rwt, tid, inv2s2);

    wait_async_then_barrier();

    const int lane  = tid & 31;
    const int wv    = tid >> 5;       // wave id 0..7 -> 16-col sub-tile
    const int ml    = lane & 15;      // A row / B col (M and N)
    const int hi    = lane >> 4;      // K-half select
    const int wbase = wv * 16;

    const float invS = 1.0f / tap_sum(rwt, lane);   // 1/Sx

    // loop-invariant bases: weight index j0 = (k0+2hi) - ml + 15 grows with k0
    const int wb0 = 2 * hi - ml + 15;          // + k0 per iteration, in [0,158]
    const int ab0 = wbase + 2 * hi;            // A column base

    v8f c = {};
    #pragma unroll
    for (int k0 = 0; k0 < TILE_K; k0 += 4) {
        // A: 16x4 data chunk; lane layout: M=lane&15, K = j + 2*(lane>=16)
        v2f a;
        a.x = lx[ml * ROWP + ab0 + k0];
        a.y = lx[ml * ROWP + ab0 + k0 + 1];
        // B: 4x16 Toeplitz band from reversed table (ascending pair)
        v2f b;
        b.x = rwt[wb0 + k0];
        b.y = rwt[wb0 + k0 + 1];
        c = __builtin_amdgcn_wmma_f32_16x16x4_f32(false, a, false, b,
                                                  (short)0, c, false, false);
    }

    // C/D layout: VGPR r -> rows r (lanes 0-15) and r+8 (lanes 16-31)
    float* op = t + (size_t)(bz * Hd + a0 + 8 * hi) * Wd + (b0 + wbase + ml);
    #pragma unroll
    for (int r = 0; r < 8; ++r) op[(size_t)r * Wd] = c[r] * invS;
}

// ---------------- Pass 2: vertical circular convolution ----------------
// Block: 256 threads, output tile 16 rows x 128 cols.
// Per wave: O = A(16xK Toeplitz weights) * B(Kx16 data columns).
__global__ __launch_bounds__(256) void blur_col_pass(const float* __restrict__ t,
                                                     const float* __restrict__ sigp,
                                                     float* __restrict__ y) {
    __shared__ __align__(16) float rwt[WTSZ];
    __shared__ __align__(16) float lt[TILE_K * COLP];  // 144 rows x 128 cols (+pad)

    const int tid = threadIdx.x;
    const int b0  = blockIdx.x * 128;   // first output column
    const int a0  = blockIdx.y * 16;    // first output row
    const int bz  = blockIdx.z;

    // stage 144 x 128 strip (row-wrapped), async global->LDS b128
    const float* tb = t + (size_t)bz * Hd * Wd;          // uniform block base
    #pragma unroll
    for (int it = 0; it < 18; ++it) {
        int idx4 = tid + it * 256;          // 144*32 float4 total
        int row  = idx4 >> 5;
        int c4   = (idx4 & 31) << 2;
        int gr   = (a0 - RAD + row) & MASKW;
        async_g2l_b128(&lt[row * COLP + c4], tb,
                       (unsigned)((gr * Wd + b0 + c4) * 4));
    }

    const float sig = sigma_from(sigp);
    const float inv2s2 = 1.0f / (2.0f * sig * sig);
    fill_taps(rwt, tid, inv2s2);

    wait_async_then_barrier();

    const int lane  = tid & 31;
    const int wv    = tid >> 5;
    const int ml    = lane & 15;
    const int hi    = lane >> 4;
    const int cbase = wv * 16;

    const float invS = 1.0f / tap_sum(rwt, lane);   // 1/Sy

    const int wb0 = 2 * hi - ml + 15;              // weight index base (+k0/iter)
    const int db0 = 2 * hi * COLP + cbase + ml;    // data base: row 2hi, col cbase+ml

    v8f c = {};
    #pragma unroll
    for (int k0 = 0; k0 < TILE_K; k0 += 4) {
        // A: 16x4 Toeplitz band from reversed table (ascending pair)
        v2f a;
        a.x = rwt[wb0 + k0];
        a.y = rwt[wb0 + k0 + 1];
        // B: 4x16 data chunk; rows k0+2hi, k0+2hi+1 at column cbase+ml
        v2f b;
        b.x = lt[db0 + k0 * COLP];
        b.y = lt[db0 + k0 * COLP + COLP];
        c = __builtin_amdgcn_wmma_f32_16x16x4_f32(false, a, false, b,
                                                  (short)0, c, false, false);
    }

    float* op = y + (size_t)(bz * Hd + a0 + 8 * hi) * Wd + (b0 + cbase + ml);
    #pragma unroll
    for (int r = 0; r < 8; ++r) op[(size_t)r * Wd] = c[r] * invS;
}

extern "C" void kernel_launch(void* const* d_in, const int* in_sizes, int n_in,
                              void* d_out, int out_size, void* d_ws, size_t ws_size,
                              hipStream_t stream) {
    const float* x    = (const float*)d_in[0];  // [8,1,2048,2048] fp32
    const float* sigp = (const float*)d_in[1];  // scalar sigma_
    float* out = (float*)d_out;
    float* tmp = (float*)d_ws;                  // 128 MiB intermediate

    dim3 blk(256);
    dim3 grd(Wd / 128, Hd / 16, NB);            // 16 x 128 x 8 blocks per pass
    blur_row_pass<<<grd, blk, 0, stream>>>(x, sigp, tmp);
    blur_col_pass<<<grd, blk, 0, stream>>>(tmp, sigp, out);
}